// gdfn_img_73194832658612
// MI455X (gfx1250) — compile-verified
//
#include <hip/hip_runtime.h>

// GDFN-style data-dependent 3x3 bilateral filter, fused single kernel.
// img: (3,2048,2048) f32, noise: (3,2050,2050) f32, out: (3,2048,2048) i32.

static constexpr int IMG_H = 2048;   // image height/width
static constexpr int PAD_H = 2050;   // padded (noise) height/width
static constexpr int TILE  = 32;     // output tile per block
static constexpr int PADW  = TILE + 6;  // padded-image tile extent (halo 2 for img_r + patch)
static constexpr int RW    = TILE + 4;  // img_r extent (halo 2)
static constexpr int MW    = TILE + 2;  // img_3d_mod extent (halo 1)
static constexpr int NTHREADS = 256;    // 8 waves of 32

__global__ __launch_bounds__(NTHREADS)
void gdfn_fused_kernel(const float* __restrict__ img,
                       const float* __restrict__ noise,
                       int* __restrict__ out) {
  __shared__ float sPad[3 * PADW * PADW];  // pad(img)+noise, origin (y0-2, x0-2) in padded coords
  __shared__ float sR[RW * RW];            // img_r, origin (y0-2, x0-2) in image coords
  __shared__ float sMod[MW * MW];          // img_3d_mod, origin (y0-1, x0-1) in image coords

  const int tid = threadIdx.x;
  const int x0 = blockIdx.x * TILE;
  const int y0 = blockIdx.y * TILE;

  // ---- Stage 0: async DMA the noise tile (padded coords) straight into LDS ----
  for (int idx = tid; idx < 3 * PADW * PADW; idx += NTHREADS) {
    const int c   = idx / (PADW * PADW);
    const int rem = idx - c * (PADW * PADW);
    const int ly  = rem / PADW;
    const int lx  = rem - ly * PADW;
    const int py  = y0 - 2 + ly;             // padded-coord row
    const int px  = x0 - 2 + lx;             // padded-coord col
    const unsigned lds_addr = (unsigned)(unsigned long long)(&sPad[idx]);
    if (py >= 0 && py < PAD_H && px >= 0 && px < PAD_H) {
      const unsigned goff = (unsigned)((c * PAD_H * PAD_H + py * PAD_H + px) * 4);
      // GVS form: mem_addr = SADDR(u64) + VADDR(i32) + IOFFSET; LDS dest addr in first VGPR.
      asm volatile("global_load_async_to_lds_b32 %0, %1, %2 offset:0"
                   :: "v"(lds_addr), "v"(goff), "s"(noise) : "memory");
    } else {
      sPad[idx] = 0.0f;  // outside padded tensor: never read, keep clean
    }
  }
  asm volatile("s_wait_asynccnt 0x0" ::: "memory");  // this wave's DMA done

  // In-place add of zero-padded img (same idx partition -> same-thread RMW, no races).
  for (int idx = tid; idx < 3 * PADW * PADW; idx += NTHREADS) {
    const int c   = idx / (PADW * PADW);
    const int rem = idx - c * (PADW * PADW);
    const int ly  = rem / PADW;
    const int lx  = rem - ly * PADW;
    const int py  = y0 - 2 + ly;
    const int px  = x0 - 2 + lx;
    const int iy  = py - 1;  // image coords
    const int ix  = px - 1;
    if (py >= 0 && py < PAD_H && px >= 0 && px < PAD_H &&
        iy >= 0 && iy < IMG_H && ix >= 0 && ix < IMG_H) {
      sPad[idx] += img[c * IMG_H * IMG_H + iy * IMG_H + ix];
    }
  }
  __syncthreads();

  // ---- Stage 1: img_r over the (TILE+4)^2 halo region ----
  for (int idx = tid; idx < RW * RW; idx += NTHREADS) {
    const int al = idx / RW;
    const int bl = idx - al * RW;
    const int a  = y0 - 2 + al;   // image coords
    const int b  = x0 - 2 + bl;
    float r = 0.0f;
    if (a >= 0 && a < IMG_H && b >= 0 && b < IMG_H) {
      float vs[3][9];
      float w3[9];
#pragma unroll
      for (int k = 0; k < 9; ++k) w3[k] = 3.402823e38f;
#pragma unroll
      for (int c = 0; c < 3; ++c) {
        float s = 0.0f, ss = 0.0f;
#pragma unroll
        for (int dy = 0; dy < 3; ++dy) {
#pragma unroll
          for (int dx = 0; dx < 3; ++dx) {
            const float v = sPad[c * PADW * PADW + (al + dy) * PADW + (bl + dx)];
            vs[c][dy * 3 + dx] = v;
            s += v;
            ss += v * v;
          }
        }
        const float m   = s * (1.0f / 9.0f);
        const float var = (ss - 9.0f * m * m) * (1.0f / 8.0f);  // ddof=1
        const float inv2v = 0.5f / var;
#pragma unroll
        for (int k = 0; k < 9; ++k) {
          const float d  = vs[c][k] - m;
          const float wc = __expf(-(d * d) * inv2v);
          w3[k] = fminf(w3[k], wc);          // min over channels
        }
      }
      float num = 0.0f, den = 0.0f;
#pragma unroll
      for (int k = 0; k < 9; ++k) {
        den += w3[k];
        num += (vs[0][k] + vs[1][k] + vs[2][k]) * w3[k];
      }
      r = num / den;
    }
    sR[idx] = r;  // zero outside image == img_r zero padding
  }
  __syncthreads();

  // ---- Stage 2: img_3d_mod over the (TILE+2)^2 halo region ----
  for (int idx = tid; idx < MW * MW; idx += NTHREADS) {
    const int vl = idx / MW;
    const int ul = idx - vl * MW;
    const int v  = y0 - 1 + vl;   // image coords
    const int u  = x0 - 1 + ul;
    float w = 0.0f;
    if (v >= 0 && v < IMG_H && u >= 0 && u < IMG_H) {
      // argmax of img_r 3x3 neighborhood; '>=' keeps the LAST tie (stable argsort, last index)
      float best = -3.402823e38f;
      int pbest = 0;
#pragma unroll
      for (int dy = 0; dy < 3; ++dy) {
#pragma unroll
        for (int dx = 0; dx < 3; ++dx) {
          const int gy = v + dy - 1, gx = u + dx - 1;
          float rv = 0.0f;
          if (gy >= 0 && gy < IMG_H && gx >= 0 && gx < IMG_H)
            rv = sR[(vl + dy) * RW + (ul + dx)];
          const int k = dy * 3 + dx;
          if (rv >= best) { best = rv; pbest = k; }
        }
      }
      const int pdy = pbest / 3;
      const int pdx = pbest - pdy * 3;
      // recompute only the selected Gaussian weight, min over channels
      w = 3.402823e38f;
#pragma unroll
      for (int c = 0; c < 3; ++c) {
        float s = 0.0f, ss = 0.0f, vp = 0.0f;
#pragma unroll
        for (int dy = 0; dy < 3; ++dy) {
#pragma unroll
          for (int dx = 0; dx < 3; ++dx) {
            const float val = sPad[c * PADW * PADW + (vl + 1 + dy) * PADW + (ul + 1 + dx)];
            s += val;
            ss += val * val;
            if (dy == pdy && dx == pdx) vp = val;
          }
        }
        const float m   = s * (1.0f / 9.0f);
        const float var = (ss - 9.0f * m * m) * (1.0f / 8.0f);
        const float d   = vp - m;
        w = fminf(w, __expf(-(d * d) * (0.5f / var)));
      }
    }
    sMod[idx] = w;  // zero outside image == mod zero padding
  }
  __syncthreads();

  // ---- Stage 3: second weighted filter + int32 store ----
  for (int idx = tid; idx < TILE * TILE; idx += NTHREADS) {
    const int ty = idx / TILE;
    const int tx = idx - ty * TILE;
    const int y = y0 + ty;
    const int x = x0 + tx;
    float me[9];
    float den = 0.0f;
#pragma unroll
    for (int dy = 0; dy < 3; ++dy) {
#pragma unroll
      for (int dx = 0; dx < 3; ++dx) {
        const float m = sMod[(ty + dy) * MW + (tx + dx)];
        me[dy * 3 + dx] = m;
        den += m;
      }
    }
    const float invden = 1.0f / den;
#pragma unroll
    for (int c = 0; c < 3; ++c) {
      float num = 0.0f;
#pragma unroll
      for (int dy = 0; dy < 3; ++dy) {
#pragma unroll
        for (int dx = 0; dx < 3; ++dx) {
          num += sPad[c * PADW * PADW + (ty + 2 + dy) * PADW + (tx + 2 + dx)] * me[dy * 3 + dx];
        }
      }
      out[c * IMG_H * IMG_H + y * IMG_H + x] = (int)(num * invden);  // trunc toward zero
    }
  }
}

extern "C" void kernel_launch(void* const* d_in, const int* in_sizes, int n_in,
                              void* d_out, int out_size, void* d_ws, size_t ws_size,
                              hipStream_t stream) {
  const float* img   = (const float*)d_in[0];
  const float* noise = (const float*)d_in[1];
  int* out = (int*)d_out;
  dim3 grid(IMG_H / TILE, IMG_H / TILE);  // 64 x 64 tiles
  gdfn_fused_kernel<<<grid, NTHREADS, 0, stream>>>(img, noise, out);
}